// ProposalTargetLayer_17300128268723
// MI455X (gfx1250) — compile-verified
//
#include <hip/hip_runtime.h>
#include <hip/hip_bf16.h>
#include <stdint.h>

// ---------------- problem constants ----------------
#define R_TOT 6000
#define R_PAD 6016   // 47 * 128
#define G_TOT 100
#define G_PAD 112    // 7 * 16
#define B_TOT 32
#define NS    128
#define POSMAX 32

typedef __attribute__((ext_vector_type(2))) float v2f;
typedef __attribute__((ext_vector_type(8))) float v8f;

// ---------------- Threefry-2x32 (JAX-compatible, 20 rounds) ----------------
__host__ __device__ inline uint32_t rotl32(uint32_t x, int r) { return (x << r) | (x >> (32 - r)); }

__host__ __device__ inline void threefry2x32(uint32_t k0, uint32_t k1, uint32_t& x0, uint32_t& x1) {
  uint32_t ks2 = k0 ^ k1 ^ 0x1BD11BDAu;
  x0 += k0; x1 += k1;
#define TF_R(r) { x0 += x1; x1 = rotl32(x1, (r)); x1 ^= x0; }
  TF_R(13) TF_R(15) TF_R(26) TF_R(6)
  x0 += k1;  x1 += ks2 + 1u;
  TF_R(17) TF_R(29) TF_R(16) TF_R(24)
  x0 += ks2; x1 += k0 + 2u;
  TF_R(13) TF_R(15) TF_R(26) TF_R(6)
  x0 += k0;  x1 += k1 + 3u;
  TF_R(17) TF_R(29) TF_R(16) TF_R(24)
  x0 += k1;  x1 += ks2 + 4u;
  TF_R(13) TF_R(15) TF_R(26) TF_R(6)
  x0 += ks2; x1 += k0 + 5u;
#undef TF_R
}

struct Keys { uint32_t k0[B_TOT]; uint32_t k1[B_TOT]; };

__device__ inline float tf_uniform(uint32_t bits) {
  // JAX: bitcast(bits>>9 | 0x3F800000) - 1  -> [0,1); then *(max-min)+min; then max(min, .)
  float f = __uint_as_float((bits >> 9) | 0x3F800000u) - 1.0f;
  float r = f * (1.0f - 1e-6f) + 1e-6f;
  return fmaxf(r, 1e-6f);
}

// LDS byte-address of a __shared__ object (generic -> addrspace(3) -> u32)
__device__ inline unsigned lds_addr_of(const void* p) {
  return (unsigned)(unsigned long long)(__attribute__((address_space(3))) const void*)p;
}

// ---------------- Kernel 0: per-ROI random priorities r[b][i] ----------------
__global__ __launch_bounds__(256) void k0_rng(Keys keys, float* __restrict__ wr) {
  const int b = blockIdx.y;
  const int c = blockIdx.x * 256 + threadIdx.x;
  if (c >= R_TOT / 2) return;                 // 3000 counter pairs -> 6000 values
  uint32_t x0 = (uint32_t)c, x1 = (uint32_t)(c + R_TOT / 2);
  threefry2x32(keys.k0[b], keys.k1[b], x0, x1);
  wr[(size_t)b * R_PAD + c]              = tf_uniform(x0);
  wr[(size_t)b * R_PAD + c + R_TOT / 2]  = tf_uniform(x1);
}

// ---------------- Kernel 1: IoU max/argmax via WMMA-assisted tiles ----------------
// grid = (47, 32), block = 256 (8 waves). Wave w handles 16 ROIs x all 112 (padded) GTs.
__global__ __launch_bounds__(256) void k1_iou(const float* __restrict__ roi,
                                              const float* __restrict__ gtb,
                                              float* __restrict__ wmax,
                                              int*   __restrict__ warg) {
  __shared__ float4 sgt[G_PAD];
  const int b = blockIdx.y;
  const int t = threadIdx.x;

  // Stage GT boxes into LDS with CDNA5 async-to-LDS DMA; zero-pad tail tiles.
  if (t < G_TOT) {
    const float* src = gtb + ((size_t)b * G_TOT + t) * 4;
    unsigned lds = lds_addr_of(&sgt[t]);
    asm volatile("global_load_async_to_lds_b128 %0, %1, off" :: "v"(lds), "v"(src) : "memory");
  } else if (t < G_PAD) {
    sgt[t] = float4{0.f, 0.f, 0.f, 0.f};
  }
  asm volatile("s_wait_asynccnt 0" ::: "memory");
  __syncthreads();

  const int wave = t >> 5;
  const int lane = t & 31;
  const bool hi  = lane >= 16;
  const int  n   = lane & 15;                       // output column (GT within tile)
  const int  base = blockIdx.x * 128 + wave * 16;   // first ROI of this wave's tile

  const float* rb = roi + (size_t)b * R_TOT * 4;
  __builtin_prefetch(rb + (size_t)base * 4, 0, 3);  // global_prefetch_b8

  // A fragment (16x4 f32): lanes 0-15 hold row M=lane with K0=area_r, K1=1; lanes 16-31 hold K2,K3 = 0.
  int aRow = base + lane;
  int aRowC = aRow < R_TOT ? aRow : (R_TOT - 1);
  float4 abox = ((const float4*)rb)[hi ? 0 : aRowC];
  float areaA = (abox.z - abox.x) * (abox.w - abox.y);
  v2f A; A.x = hi ? 0.f : areaA; A.y = hi ? 0.f : 1.0f;

  // ROI boxes for this lane's 8 output rows (M = v + 8*hi)
  const int r0 = base + (hi ? 8 : 0);
  float4 rbox[8];
#pragma unroll
  for (int v = 0; v < 8; ++v) {
    int rr = r0 + v; rr = rr < R_TOT ? rr : (R_TOT - 1);
    rbox[v] = ((const float4*)rb)[rr];
  }

  float bestV[8]; int bestI[8];
#pragma unroll
  for (int v = 0; v < 8; ++v) { bestV[v] = -1.0f; bestI[v] = 0; }

#pragma unroll
  for (int tile = 0; tile < G_PAD / 16; ++tile) {
    const int gIdx = tile * 16 + n;
    float4 g = sgt[gIdx];
    float areaG = (g.z - g.x) * (g.w - g.y);

    // B fragment (4x16 f32): lanes 0-15: K0 row = 1, K1 row = area_g; lanes 16-31: K2,K3 rows = 0.
    v2f Bf; Bf.x = hi ? 0.f : 1.0f; Bf.y = hi ? 0.f : areaG;

    // D[m][n] = area_r[m] + area_g[n]  (exact f32, same rounding as the reference add)
    v8f D;
#if __has_builtin(__builtin_amdgcn_wmma_f32_16x16x4_f32)
    v8f C = {0.f, 0.f, 0.f, 0.f, 0.f, 0.f, 0.f, 0.f};
    D = __builtin_amdgcn_wmma_f32_16x16x4_f32(false, A, false, Bf, (short)0, C, false, false);
#else
#pragma unroll
    for (int v = 0; v < 8; ++v)
      D[v] = (rbox[v].z - rbox[v].x) * (rbox[v].w - rbox[v].y) + areaG;
#endif

#pragma unroll
    for (int v = 0; v < 8; ++v) {
      float ix1 = fmaxf(rbox[v].x, g.x);
      float iy1 = fmaxf(rbox[v].y, g.y);
      float ix2 = fminf(rbox[v].z, g.z);
      float iy2 = fminf(rbox[v].w, g.w);
      float iw = fmaxf(ix2 - ix1, 0.f);
      float ih = fmaxf(iy2 - iy1, 0.f);
      float inter = iw * ih;
      float iou = inter / (D[v] - inter);
      if (iou > bestV[v]) { bestV[v] = iou; bestI[v] = gIdx; }  // ascending gIdx keeps first-max
    }
  }

  // Reduce max/argmax across the 16 lanes of each half-wave (xor 1,2,4,8 stays within a half).
#pragma unroll
  for (int v = 0; v < 8; ++v) {
    float bv = bestV[v]; int bi = bestI[v];
#pragma unroll
    for (int m = 1; m < 16; m <<= 1) {
      float ov = __shfl_xor(bv, m, 32);
      int   oi = __shfl_xor(bi, m, 32);
      if (ov > bv || (ov == bv && oi < bi)) { bv = ov; bi = oi; }
    }
    bestV[v] = bv; bestI[v] = bi;
  }

  if (n == 0) {  // lanes 0 and 16 write rows r0..r0+7
#pragma unroll
    for (int v = 0; v < 8; ++v) {
      int i = r0 + v;
      bool ok = i < R_TOT;
      wmax[(size_t)b * R_PAD + i] = ok ? bestV[v] : -1.0f;
      warg[(size_t)b * R_PAD + i] = ok ? bestI[v] : 0;
    }
  }
}

// ---------------- Kernel 2: per-batch rank-based sampling + outputs ----------------
// grid = (32), block = 1024
__global__ __launch_bounds__(1024) void k2_select(const float* __restrict__ roi,
                                                  const float* __restrict__ gtb,
                                                  const int*   __restrict__ labels,
                                                  const float* __restrict__ wr,
                                                  const float* __restrict__ wmax,
                                                  const int*   __restrict__ warg,
                                                  float* __restrict__ out) {
  __shared__ float s_r[R_PAD];
  __shared__ unsigned char s_m[R_PAD];  // 0=excluded, 1=positive, 2=negative
  __shared__ int s_keep[NS];
  __shared__ int s_pos[NS];
  __shared__ int s_npos;

  const int b = blockIdx.x;
  const int t = threadIdx.x;

  if (t == 0) s_npos = 0;
  if (t < NS) { s_keep[t] = -1; s_pos[t] = 0; }
  __syncthreads();

  int localPos = 0;
  for (int i = t; i < R_PAD; i += 1024) {
    float mi = wmax[(size_t)b * R_PAD + i];
    float rv = wr[(size_t)b * R_PAD + i];
    unsigned char m = 0;
    if (mi >= 0.5f) { m = 1; localPos++; }
    else if (mi >= 0.0f) m = 2;                 // NEG_LO <= iou < NEG_HI
    s_r[i] = rv;
    s_m[i] = m;
  }
  if (localPos) atomicAdd(&s_npos, localPos);
  __syncthreads();

  const int nSelPos = min(s_npos, POSMAX);      // = sum(is_sel_pos) in the reference
  const int negQuota = NS - nSelPos;

  // Rank by counting (stable tie-break on index == argsort(argsort(-x)))
  for (int i = t; i < R_PAD; i += 1024) {
    const unsigned char m = s_m[i];
    if (m == 0) continue;
    const float ri = s_r[i];
    int rank = 0;
    for (int j = 0; j < R_PAD; ++j) {
      if (s_m[j] == m) {
        const float rj = s_r[j];
        rank += (rj > ri) || (rj == ri && j < i);
      }
    }
    if (m == 1) {
      if (rank < POSMAX) { s_keep[rank] = i; s_pos[rank] = 1; }
    } else {
      if (rank < negQuota) { s_keep[nSelPos + rank] = i; s_pos[nSelPos + rank] = 0; }
    }
  }
  __syncthreads();

  if (t < NS) {
    const size_t slotG = (size_t)b * NS + t;
    float* outRoi = out;                                     // [32,128,4]
    float* outLab = out + (size_t)B_TOT * NS * 4;            // [32,128]
    float* outDel = out + (size_t)B_TOT * NS * 5;            // [32,128,4]

    float4 sroi = {0.f, 0.f, 0.f, 0.f};
    float4 del  = {0.f, 0.f, 0.f, 0.f};
    float  lab  = 0.f;

    const int i = s_keep[t];
    if (i >= 0) {
      const float4 r4 = ((const float4*)(roi + (size_t)b * R_TOT * 4))[i];
      const int a = warg[(size_t)b * R_PAD + i];
      const float4 g4 = ((const float4*)(gtb + (size_t)b * G_TOT * 4))[a];
      if (s_pos[t]) lab = (float)labels[b * G_TOT + a];
      sroi = r4;
      const float EPSF = 1.1920928955078125e-07f;   // np.finfo(float32).eps
      float w  = fmaxf(r4.z - r4.x, EPSF);
      float h  = fmaxf(r4.w - r4.y, EPSF);
      float cx = r4.x + 0.5f * (r4.z - r4.x);
      float cy = r4.y + 0.5f * (r4.w - r4.y);
      float bw = g4.z - g4.x;
      float bh = g4.w - g4.y;
      float bcx = g4.x + 0.5f * bw;
      float bcy = g4.y + 0.5f * bh;
      del.x = (bcx - cx) / w;
      del.y = (bcy - cy) / h;
      del.z = logf(bw / w);
      del.w = logf(bh / h);
    }
    ((float4*)outRoi)[slotG] = sroi;
    outLab[slotG] = lab;
    ((float4*)outDel)[slotG] = del;
  }
}

// ---------------- host launcher ----------------
extern "C" void kernel_launch(void* const* d_in, const int* in_sizes, int n_in,
                              void* d_out, int out_size, void* d_ws, size_t ws_size,
                              hipStream_t stream) {
  const float* roi    = (const float*)d_in[0];   // [32,6000,4]
  const float* gtb    = (const float*)d_in[1];   // [32,100,4]
  const int*   labels = (const int*)d_in[2];     // [32,100]
  // d_in[3] (image) is unused by the math.

  float* wr   = (float*)d_ws;                    // [32, 6016] priorities
  float* wmax = wr + (size_t)B_TOT * R_PAD;      // [32, 6016] max IoU (-1 = pad)
  int*   warg = (int*)(wmax + (size_t)B_TOT * R_PAD); // [32, 6016] argmax

  // Reproduce jax.random.split(jax.random.key(42), 32) on the host (Threefry-2x32).
  Keys keys;
  {
    uint32_t outw[64];
    for (int i = 0; i < 32; ++i) {
      uint32_t x0 = (uint32_t)i, x1 = (uint32_t)(32 + i);   // counters iota(64) split in halves
      threefry2x32(0u, 42u, x0, x1);
      outw[i] = x0; outw[32 + i] = x1;
    }
    for (int b = 0; b < B_TOT; ++b) { keys.k0[b] = outw[2 * b]; keys.k1[b] = outw[2 * b + 1]; }
  }

  k0_rng<<<dim3((R_TOT / 2 + 255) / 256, B_TOT), dim3(256), 0, stream>>>(keys, wr);
  k1_iou<<<dim3(R_PAD / 128, B_TOT), dim3(256), 0, stream>>>(roi, gtb, wmax, warg);
  k2_select<<<dim3(B_TOT), dim3(1024), 0, stream>>>(roi, gtb, labels, wr, wmax, warg, (float*)d_out);
}